// PyrUpBicubic2d_21122649162262
// MI455X (gfx1250) — compile-verified
//
#include <hip/hip_runtime.h>
#include <hip/hip_bf16.h>
#include <stdint.h>

// ---------------------------------------------------------------------------
// PyrUpBicubic2d (a = -0.75) 2x upsample, f32, NCHW (8,128,128,128)->(8,128,256,256)
// Memory-bound (335 MB @ 23.3 TB/s ~= 14 us floor). CDNA5 paths used:
//  - Tensor Data Mover (tensor_load_to_lds + s_wait_tensorcnt) stages input
//    rows into LDS with hardware row padding (stride 132 floats) reserving
//    the replicate-halo column slots.
//  - Non-temporal global_store_b128 for the 268 MB write-once output stream
//    (output > 192 MB L2; keep the reused input rows resident instead).
// Compute: separable 4-tap f32 VALU, vertical-first (FMA-minimal).
// ---------------------------------------------------------------------------

typedef unsigned int u32x4 __attribute__((ext_vector_type(4)));
typedef int          i32x4 __attribute__((ext_vector_type(4)));
typedef int          i32x8 __attribute__((ext_vector_type(8)));
typedef float        f32x2 __attribute__((ext_vector_type(2)));
typedef float        f32x4 __attribute__((ext_vector_type(4)));

#define IH 128
#define IW 128
#define STRIP_ROWS 32      // output rows per block
#define MAX_TROWS 20       // input rows staged per block (<= 20)
#define TILE_W 132         // 2 halo + 128 data + 2 halo  (matches TDM pad 4 DW / 128 DW)

#ifndef __has_builtin
#define __has_builtin(x) 0
#endif
#if __has_builtin(__builtin_amdgcn_tensor_load_to_lds)
#define HAVE_TDM 1
#else
#define HAVE_TDM 0
#endif

__device__ __forceinline__ int iclamp(int v, int lo, int hi) {
  return v < lo ? lo : (v > hi ? hi : v);
}

__global__ __launch_bounds__(256)
void pyrup_bicubic_kernel(const float* __restrict__ src, float* __restrict__ dst)
{
  __shared__ float tile[MAX_TROWS][TILE_W];

  const int tid   = threadIdx.x;
  const int plane = blockIdx.x >> 3;      // 8 strips of 32 output rows per plane
  const int strip = blockIdx.x & 7;
  const int y0    = strip * STRIP_ROWS;   // first output row of this strip
  const int I0    = y0 >> 1;              // first conv row index i
  const int rlo   = iclamp(I0 - 2, 0, IH - 1);
  const int rhi   = iclamp(I0 + 17, 0, IH - 1);
  const int nrows = rhi - rlo + 1;        // <= MAX_TROWS

  // ---- Stage input rows [rlo, rhi] (full width) into LDS via the TDM ----
#if HAVE_TDM
  if (tid < 32) {  // wave 0 issues the descriptor; TDM ignores EXEC
    // lds_addr: byte offset of tile[0][2] (flat shared addr low 32 bits == LDS offset)
    const unsigned ldsOff =
        (unsigned)(unsigned long long)(uintptr_t)&tile[0][0] + 8u;
    const unsigned long long ga =
        (unsigned long long)(uintptr_t)(src + ((size_t)plane * IH + rlo) * IW);
    const unsigned nelem = (unsigned)nrows * (unsigned)IW;  // <= 2560 f32

    // D# group 0: count=1 | lds_addr | global_addr[56:0] | type=2
    u32x4 g0 = { 1u,
                 ldsOff,
                 (unsigned)ga,
                 (unsigned)((ga >> 32) & 0x01FFFFFFu) | (2u << 30) };

    // D# group 1:
    //  [0] data_size=2 (4B), pad_enable=1, pad_interval=6 (128 DW), pad_amount=3 (4 DW)
    //  tensor_dim0 = nelem (bits 79:48), tensor_dim1 = 1 (bits 111:80)
    //  tile_dim0 = nelem (bits 127:112), tile_dim1 = 0 -> 1-D tile
    //  tensor_dim0_stride = nelem (bits 207:160)
    i32x8 g1 = { (int)((2u << 16) | (1u << 20) | (6u << 22) | (3u << 25)),
                 (int)((nelem & 0xFFFFu) << 16),
                 (int)(((nelem >> 16) & 0xFFFFu) | (1u << 16)),
                 (int)((nelem & 0xFFFFu) << 16),
                 0,
                 (int)nelem,
                 0, 0 };
    i32x4 z4 = { 0, 0, 0, 0 };
#if defined(__clang_major__) && (__clang_major__ >= 23)
    i32x8 z8 = { 0, 0, 0, 0, 0, 0, 0, 0 };
    __builtin_amdgcn_tensor_load_to_lds(g0, g1, z4, z4, z8, 0);
#else
    __builtin_amdgcn_tensor_load_to_lds(g0, g1, z4, z4, 0);
#endif
    __builtin_amdgcn_s_wait_tensorcnt(0);  // issuing wave waits; others gated by barrier
  }
#else
  for (int e = tid; e < nrows * IW; e += 256) {
    tile[e / IW][2 + (e % IW)] = src[((size_t)plane * IH + rlo) * IW + e];
  }
#endif
  __syncthreads();

  // ---- Fill 2-column replicate halos (edge padding) ----
  if (tid < nrows) {
    const float lv = tile[tid][2];
    const float rv = tile[tid][129];
    tile[tid][0]   = lv;
    tile[tid][1]   = lv;
    tile[tid][130] = rv;
    tile[tid][131] = rv;
  }
  __syncthreads();

  // ---- Separable 4-tap compute: each thread -> 4 contiguous output pixels ----
  const float we0 = -0.10546875f, we1 = 0.87890625f,
              we2 =  0.26171875f, we3 = -0.03515625f;

  const int lane = tid & 63;       // output quad column: x = 4*lane .. 4*lane+3
  const int ty   = tid >> 6;       // 4 row-groups
  const int cx   = 2 * lane;       // first LDS tap column (image col 2*lane - 2)
  float* const outPlane = dst + (size_t)plane * (2 * IH) * (2 * IW);

  for (int yy = ty; yy < STRIP_ROWS; yy += 4) {
    const int y = y0 + yy;
    const int i = (y + 1) >> 1;
    const int p = (y + 1) & 1;     // 0 -> we row taps, 1 -> wo (reversed) row taps

    float wy0, wy1, wy2, wy3;
    if (p) { wy0 = we3; wy1 = we2; wy2 = we1; wy3 = we0; }
    else   { wy0 = we0; wy1 = we1; wy2 = we2; wy3 = we3; }

    const int r0 = iclamp(i - 2, 0, IH - 1) - rlo;
    const int r1 = iclamp(i - 1, 0, IH - 1) - rlo;
    const int r2 = iclamp(i    , 0, IH - 1) - rlo;
    const int r3 = iclamp(i + 1, 0, IH - 1) - rlo;

    // Vertical pass over 6 tap columns (bank-conflict-free ds_load_b64 x3 per row)
    f32x2 a0 = *(const f32x2*)&tile[r0][cx];
    f32x2 a1 = *(const f32x2*)&tile[r0][cx + 2];
    f32x2 a2 = *(const f32x2*)&tile[r0][cx + 4];
    float v0 = wy0 * a0.x, v1 = wy0 * a0.y, v2 = wy0 * a1.x,
          v3 = wy0 * a1.y, v4 = wy0 * a2.x, v5 = wy0 * a2.y;

    a0 = *(const f32x2*)&tile[r1][cx];
    a1 = *(const f32x2*)&tile[r1][cx + 2];
    a2 = *(const f32x2*)&tile[r1][cx + 4];
    v0 = fmaf(wy1, a0.x, v0); v1 = fmaf(wy1, a0.y, v1);
    v2 = fmaf(wy1, a1.x, v2); v3 = fmaf(wy1, a1.y, v3);
    v4 = fmaf(wy1, a2.x, v4); v5 = fmaf(wy1, a2.y, v5);

    a0 = *(const f32x2*)&tile[r2][cx];
    a1 = *(const f32x2*)&tile[r2][cx + 2];
    a2 = *(const f32x2*)&tile[r2][cx + 4];
    v0 = fmaf(wy2, a0.x, v0); v1 = fmaf(wy2, a0.y, v1);
    v2 = fmaf(wy2, a1.x, v2); v3 = fmaf(wy2, a1.y, v3);
    v4 = fmaf(wy2, a2.x, v4); v5 = fmaf(wy2, a2.y, v5);

    a0 = *(const f32x2*)&tile[r3][cx];
    a1 = *(const f32x2*)&tile[r3][cx + 2];
    a2 = *(const f32x2*)&tile[r3][cx + 4];
    v0 = fmaf(wy3, a0.x, v0); v1 = fmaf(wy3, a0.y, v1);
    v2 = fmaf(wy3, a1.x, v2); v3 = fmaf(wy3, a1.y, v3);
    v4 = fmaf(wy3, a2.x, v4); v5 = fmaf(wy3, a2.y, v5);

    // Horizontal pass: phases x = even (wo = reversed we) / odd (we)
    f32x4 o;
    o.x = fmaf(we3, v0, fmaf(we2, v1, fmaf(we1, v2, we0 * v3)));
    o.y = fmaf(we0, v1, fmaf(we1, v2, fmaf(we2, v3, we3 * v4)));
    o.z = fmaf(we3, v1, fmaf(we2, v2, fmaf(we1, v3, we0 * v4)));
    o.w = fmaf(we0, v2, fmaf(we1, v3, fmaf(we2, v4, we3 * v5)));

    // Write-once 268 MB stream: non-temporal so it doesn't evict reused input
    // rows from the 192 MB L2 (emits global_store_b128 with NT temporal hint).
    __builtin_nontemporal_store(o, (f32x4*)&outPlane[(size_t)y * (2 * IW) + 4 * lane]);
  }
}

extern "C" void kernel_launch(void* const* d_in, const int* in_sizes, int n_in,
                              void* d_out, int out_size, void* d_ws, size_t ws_size,
                              hipStream_t stream) {
  const float* x   = (const float*)d_in[0];
  float*       out = (float*)d_out;
  const int planes = in_sizes[0] / (IH * IW);   // n*c = 1024 for the reference shapes
  dim3 grid((unsigned)(planes * 8));            // 8 strips of 32 output rows per plane
  dim3 block(256);
  pyrup_bicubic_kernel<<<grid, block, 0, stream>>>(x, out);
}